// PlaneProjection2_24773371364013
// MI455X (gfx1250) — compile-verified
//
#include <hip/hip_runtime.h>
#include <math.h>

typedef __attribute__((ext_vector_type(2))) float v2f;
typedef __attribute__((ext_vector_type(8))) float v8f;

#define NDIM 512
#define EPSF 1e-12f

// Byte-free (element) offsets of the 4 B-operand streams relative to the
// nn0 "lo" row pointer. pproj is laid out immediately after nn0 in the
// workspace, so all four fit in the 24-bit signed immediate offset field.
#define OFF_HI (16 * NDIM)          //  +32 KB : nn0 hi sub-tile
#define OFF_PP (NDIM * NDIM)        //  +1 MB  : pproj lo sub-tile
#define OFF_PPHI (NDIM * NDIM + 16 * NDIM)

// ---------------------------------------------------------------------------
// Kernel 1: inverse column norms of `normal` (normalize over dim 0).
// ---------------------------------------------------------------------------
__global__ void pp2_colnorm_kernel(const float* __restrict__ normal,
                                   float* __restrict__ cninv) {
    int i = blockIdx.x * blockDim.x + threadIdx.x;
    if (i >= NDIM) return;
    float s = 0.0f;
    #pragma unroll 8
    for (int o = 0; o < NDIM; ++o) {
        float v = normal[o * NDIM + i];
        s += v * v;
    }
    cninv[i] = 1.0f / fmaxf(sqrtf(s), EPSF);
}

// ---------------------------------------------------------------------------
// Kernel 2: per-row-o preprocessing. One block (256 threads) per row.
// Produces nn0 row, plane_proj row, d_o = nn0·pproj, g_o = ||nn0||^2.
// ---------------------------------------------------------------------------
__global__ void pp2_rowprep_kernel(const float* __restrict__ normal,
                                   const float* __restrict__ plane,
                                   const float* __restrict__ cninv,
                                   float* __restrict__ nn0,
                                   float* __restrict__ pproj,
                                   float* __restrict__ dvec,
                                   float* __restrict__ gvec) {
    const int o = blockIdx.x;
    const int t = threadIdx.x;
    const float* nr = normal + o * NDIM;
    const float* pr = plane + o * NDIM;

    float sp = 0.f, sn = 0.f, spn = 0.f, g = 0.f, t1 = 0.f, t2 = 0.f;
    for (int i = t; i < NDIM; i += 256) {
        float n = nr[i], p = pr[i], c = cninv[i];
        float n0 = n * c;
        sp  += p * p;
        sn  += n * n;
        spn += p * n;
        g   += n0 * n0;
        t1  += n0 * p;
        t2  += n0 * n;
        nn0[o * NDIM + i] = n0;
    }

    __shared__ float red[6][256];
    red[0][t] = sp; red[1][t] = sn; red[2][t] = spn;
    red[3][t] = g;  red[4][t] = t1; red[5][t] = t2;
    __syncthreads();
    for (int s = 128; s > 0; s >>= 1) {
        if (t < s) {
            #pragma unroll
            for (int j = 0; j < 6; ++j) red[j][t] += red[j][t + s];
        }
        __syncthreads();
    }
    float Sp = red[0][0], Sn = red[1][0], Spn = red[2][0];
    float G  = red[3][0], T1 = red[4][0], T2 = red[5][0];

    float np  = fmaxf(sqrtf(Sp), EPSF);
    float nn1 = fmaxf(sqrtf(Sn), EPSF);
    float ct  = Spn / (np * nn1);
    float qq  = fmaxf(Sp - 2.0f * ct * Spn + ct * ct * Sn, 0.0f);
    float rinv = 1.0f / fmaxf(sqrtf(qq), EPSF);

    if (t == 0) {
        dvec[o] = rinv * (T1 - ct * T2);
        gvec[o] = G;
    }
    for (int i = t; i < NDIM; i += 256) {
        pproj[o * NDIM + i] = (pr[i] - nr[i] * ct) * rinv;
    }
}

// ---------------------------------------------------------------------------
// Kernel 3: s_b = ||x[b,:]||^2. One block per row b.
// ---------------------------------------------------------------------------
__global__ void pp2_xnorm_kernel(const float* __restrict__ x,
                                 float* __restrict__ svec) {
    const int b = blockIdx.x;
    const int t = threadIdx.x;
    float s = 0.0f;
    for (int i = t; i < NDIM; i += 256) {
        float v = x[b * NDIM + i];
        s += v * v;
    }
    __shared__ float red[256];
    red[t] = s;
    __syncthreads();
    for (int w = 128; w > 0; w >>= 1) {
        if (t < w) red[t] += red[t + w];
        __syncthreads();
    }
    if (t == 0) svec[b] = red[0];
}

// ---------------------------------------------------------------------------
// Kernel 4: dual fp32 WMMA GEMM + fused epilogue.
// Software-pipelined (prefetch distance 1): fragments for step k+4 are
// issued before the 4 v_wmma_f32_16x16x4_f32 of step k, so matrix ops
// overlap with in-flight loads instead of s_wait_loadcnt 0 per WMMA.
// All 4 B streams share one address register via immediate offsets.
// ---------------------------------------------------------------------------
__global__ __launch_bounds__(128) void pp2_wmma_gemm_kernel(
    const float* __restrict__ x,
    const float* __restrict__ nn0,   // pproj must sit at nn0 + NDIM*NDIM
    const float* __restrict__ dvec,
    const float* __restrict__ gvec,
    const float* __restrict__ svec,
    float* __restrict__ out) {
    const int wave = blockIdx.x * (blockDim.x >> 5) + (threadIdx.x >> 5);
    const int mt = wave >> 4;     // 0..31  (M tile index)
    const int nt = wave & 15;     // 0..15  (N pair-tile index)
    const int tm = mt * 16;
    const int tn = nt * 32;

    const int lane = threadIdx.x & 31;
    const int half = lane >> 4;   // 0 or 1
    const int l = lane & 15;

    const float* ap = x   + (tm + l) * NDIM + 2 * half;
    const float* bp = nn0 + (tn + l) * NDIM + 2 * half;

    v8f c1lo = {}, c1hi = {}, c2lo = {}, c2hi = {};

    // Prologue: fragments for k = 0.
    v2f a  = *(const v2f*)(ap);
    v2f p0 = *(const v2f*)(bp);
    v2f p1 = *(const v2f*)(bp + OFF_HI);
    v2f q0 = *(const v2f*)(bp + OFF_PP);
    v2f q1 = *(const v2f*)(bp + OFF_PPHI);

    #pragma unroll 4
    for (int k = 0; k < NDIM - 4; k += 4) {
        // Prefetch step k+4 before consuming step k.
        v2f an  = *(const v2f*)(ap + k + 4);
        v2f p0n = *(const v2f*)(bp + k + 4);
        v2f p1n = *(const v2f*)(bp + k + 4 + OFF_HI);
        v2f q0n = *(const v2f*)(bp + k + 4 + OFF_PP);
        v2f q1n = *(const v2f*)(bp + k + 4 + OFF_PPHI);

        c1lo = __builtin_amdgcn_wmma_f32_16x16x4_f32(false, a, false, p0,
                                                     (short)0, c1lo, false, false);
        c1hi = __builtin_amdgcn_wmma_f32_16x16x4_f32(false, a, false, p1,
                                                     (short)0, c1hi, false, false);
        c2lo = __builtin_amdgcn_wmma_f32_16x16x4_f32(false, a, false, q0,
                                                     (short)0, c2lo, false, false);
        c2hi = __builtin_amdgcn_wmma_f32_16x16x4_f32(false, a, false, q1,
                                                     (short)0, c2hi, false, false);

        a = an; p0 = p0n; p1 = p1n; q0 = q0n; q1 = q1n;
    }

    // Epilogue k-step (no prefetch).
    c1lo = __builtin_amdgcn_wmma_f32_16x16x4_f32(false, a, false, p0,
                                                 (short)0, c1lo, false, false);
    c1hi = __builtin_amdgcn_wmma_f32_16x16x4_f32(false, a, false, p1,
                                                 (short)0, c1hi, false, false);
    c2lo = __builtin_amdgcn_wmma_f32_16x16x4_f32(false, a, false, q0,
                                                 (short)0, c2lo, false, false);
    c2hi = __builtin_amdgcn_wmma_f32_16x16x4_f32(false, a, false, q1,
                                                 (short)0, c2hi, false, false);

    const float dlo = dvec[tn + l];
    const float dhi = dvec[tn + 16 + l];
    const float glo = gvec[tn + l] - 2.0f;
    const float ghi = gvec[tn + 16 + l] - 2.0f;

    #pragma unroll
    for (int r = 0; r < 8; ++r) {
        const int row = tm + r + half * 8;
        const float sb = svec[row];

        float c1 = c1lo[r], c2 = c2lo[r];
        float den = fmaxf(sqrtf(fmaxf(sb + c1 * c1 * glo, 0.0f)), EPSF);
        out[row * NDIM + tn + l] = (c2 - c1 * dlo) / den;

        c1 = c1hi[r]; c2 = c2hi[r];
        den = fmaxf(sqrtf(fmaxf(sb + c1 * c1 * ghi, 0.0f)), EPSF);
        out[row * NDIM + tn + 16 + l] = (c2 - c1 * dhi) / den;
    }
}

// ---------------------------------------------------------------------------
// Launch
// ---------------------------------------------------------------------------
extern "C" void kernel_launch(void* const* d_in, const int* in_sizes, int n_in,
                              void* d_out, int out_size, void* d_ws, size_t ws_size,
                              hipStream_t stream) {
    const float* x      = (const float*)d_in[0];
    const float* normal = (const float*)d_in[1];
    const float* plane  = (const float*)d_in[2];
    float* out = (float*)d_out;

    float* ws    = (float*)d_ws;
    float* nn0   = ws;                       // 512*512  (pproj MUST follow)
    float* pproj = ws + NDIM * NDIM;         // 512*512
    float* cninv = ws + 2 * NDIM * NDIM;     // 512
    float* dvec  = cninv + NDIM;             // 512
    float* gvec  = dvec + NDIM;              // 512
    float* svec  = gvec + NDIM;              // 512

    pp2_colnorm_kernel<<<(NDIM + 255) / 256, 256, 0, stream>>>(normal, cninv);
    pp2_rowprep_kernel<<<NDIM, 256, 0, stream>>>(normal, plane, cninv,
                                                 nn0, pproj, dvec, gvec);
    pp2_xnorm_kernel<<<NDIM, 256, 0, stream>>>(x, svec);

    // 512 waves total: 32 M-tiles x 16 N-pair-tiles; 4 waves per 128-thread block.
    pp2_wmma_gemm_kernel<<<128, 128, 0, stream>>>(x, nn0,
                                                  dvec, gvec, svec, out);
}